// GAdapter_64063732187633
// MI455X (gfx1250) — compile-verified
//
#include <hip/hip_runtime.h>
#include <hip/hip_bf16.h>

typedef float v2f __attribute__((ext_vector_type(2)));
typedef float v8f __attribute__((ext_vector_type(8)));

#define H 128
#define LN_EPS 1e-5f

static __device__ __forceinline__ v8f wmma_f32(v2f a, v2f b, v8f c) {
    // D = A(16x4 f32) * B(4x16 f32) + C(16x16 f32)
    return __builtin_amdgcn_wmma_f32_16x16x4_f32(
        /*neg_a=*/false, a, /*neg_b=*/false, b,
        /*c_mod=*/(short)0, c, /*reuse_a=*/false, /*reuse_b=*/false);
}

// ---------------- Kernel 1: pre-LN (wave per row) + zero agg ----------------
__global__ __launch_bounds__(256) void k_preln(const float* __restrict__ x,
                                               const float* __restrict__ g,
                                               const float* __restrict__ b,
                                               float* __restrict__ h,
                                               float* __restrict__ agg) {
    const int row  = blockIdx.x * 8 + (threadIdx.x >> 5);
    const int lane = threadIdx.x & 31;
    const float4 v = ((const float4*)(x + (size_t)row * H))[lane];
    float s  = v.x + v.y + v.z + v.w;
    float s2 = v.x * v.x + v.y * v.y + v.z * v.z + v.w * v.w;
#pragma unroll
    for (int off = 16; off > 0; off >>= 1) {
        s  += __shfl_xor(s,  off, 32);
        s2 += __shfl_xor(s2, off, 32);
    }
    const float mean = s * (1.0f / H);
    const float var  = s2 * (1.0f / H) - mean * mean;
    const float rs   = rsqrtf(var + LN_EPS);
    const float4 gg = ((const float4*)g)[lane];
    const float4 bb = ((const float4*)b)[lane];
    float4 o;
    o.x = (v.x - mean) * rs * gg.x + bb.x;
    o.y = (v.y - mean) * rs * gg.y + bb.y;
    o.z = (v.z - mean) * rs * gg.z + bb.z;
    o.w = (v.w - mean) * rs * gg.w + bb.w;
    ((float4*)(h + (size_t)row * H))[lane] = o;
    ((float4*)(agg + (size_t)row * H))[lane] = make_float4(0.f, 0.f, 0.f, 0.f);
}

// ---------------- Kernel 2: edge scatter  agg[r,:] += h[c,:] ----------------
__global__ __launch_bounds__(256) void k_scatter(const int* __restrict__ ei,
                                                 const float* __restrict__ h,
                                                 float* agg, int E) {
    const unsigned t = blockIdx.x * 256u + threadIdx.x;
    const int e    = (int)(t >> 5);
    const int lane = (int)(t & 31u);
    if (e >= E) return;
    const int r = ei[e];
    const int c = ei[E + e];
    const float4 v = ((const float4*)(h + (size_t)c * H))[lane];
    float* dst = agg + (size_t)r * H + lane * 4;
    __hip_atomic_fetch_add(dst + 0, v.x, __ATOMIC_RELAXED, __HIP_MEMORY_SCOPE_AGENT);
    __hip_atomic_fetch_add(dst + 1, v.y, __ATOMIC_RELAXED, __HIP_MEMORY_SCOPE_AGENT);
    __hip_atomic_fetch_add(dst + 2, v.z, __ATOMIC_RELAXED, __HIP_MEMORY_SCOPE_AGENT);
    __hip_atomic_fetch_add(dst + 3, v.w, __ATOMIC_RELAXED, __HIP_MEMORY_SCOPE_AGENT);
}

// ------- Kernel 3: fused down-proj -> up-proj -> relu+res -> post-LN --------
// One wave (32 lanes) owns a 16-row tile. 4 waves per block.
__global__ __launch_bounds__(128) void k_mlp(const float* agg,           // [N,128] (aliases out)
                                             const float* __restrict__ hres, // [N,128]
                                             const float* __restrict__ dw,   // [32,128]
                                             const float* __restrict__ db,   // [32]
                                             const float* __restrict__ uw,   // [128,32]
                                             const float* __restrict__ ub,   // [128]
                                             const float* __restrict__ pg,   // [128]
                                             const float* __restrict__ pb,   // [128]
                                             float* out) {
    __shared__ __align__(16) float lds_z[4 * 16 * 32];   // z tiles, per wave
    __shared__ __align__(16) float lds_o[4 * 16 * 132];  // padded output tiles

    const int wid  = threadIdx.x >> 5;
    const int lane = threadIdx.x & 31;
    const int m0   = (blockIdx.x * 4 + wid) * 16;
    const int ln16 = lane & 15;   // M for A-layout / N for B,C layouts
    const int kh   = lane >> 4;   // lane-half -> K offset (A/B), +8 rows (C/D)

    float* zt = lds_z + wid * (16 * 32);
    float* ot = lds_o + wid * (16 * 132);

    // ---- GEMM1: z[16,32] = agg[16,128] @ dw^T + db ----
    v8f acc0 = {}, acc1 = {};
    const float2* arow = (const float2*)(agg + (size_t)(m0 + ln16) * H);
    const float2* b0row = (const float2*)(dw + (size_t)ln16 * H);
    const float2* b1row = (const float2*)(dw + (size_t)(16 + ln16) * H);
#pragma unroll
    for (int k0 = 0; k0 < 128; k0 += 4) {
        const int ki = (k0 >> 1) + kh;          // element pair index: K = k0 + 2*kh
        const float2 av = arow[ki];
        const float2 b0 = b0row[ki];
        const float2 b1 = b1row[ki];
        v2f a;  a.x = av.x;  a.y = av.y;
        v2f vb0; vb0.x = b0.x; vb0.y = b0.y;
        v2f vb1; vb1.x = b1.x; vb1.y = b1.y;
        acc0 = wmma_f32(a, vb0, acc0);
        acc1 = wmma_f32(a, vb1, acc1);
    }
    const float bj0 = db[ln16], bj1 = db[16 + ln16];
#pragma unroll
    for (int r = 0; r < 8; ++r) { acc0[r] += bj0; acc1[r] += bj1; }

    // C-layout -> LDS z tile [16][32]
#pragma unroll
    for (int r = 0; r < 8; ++r) {
        const int M = r + 8 * kh;
        zt[M * 32 + ln16]      = acc0[r];
        zt[M * 32 + 16 + ln16] = acc1[r];
    }
    asm volatile("s_wait_dscnt 0" ::: "memory");   // wave-local LDS RAW fence

    // ---- GEMM2: zu[16,128] = z[16,32] @ uw^T + ub ----
    v8f acc[8] = {};
    const float2* a2row = (const float2*)(zt + ln16 * 32);
#pragma unroll
    for (int k0 = 0; k0 < 32; k0 += 4) {
        const int ki = (k0 >> 1) + kh;
        const float2 av = a2row[ki];
        v2f a; a.x = av.x; a.y = av.y;
#pragma unroll
        for (int t = 0; t < 8; ++t) {
            const float2 bv = ((const float2*)(uw + (size_t)(t * 16 + ln16) * 32))[ki];
            v2f b; b.x = bv.x; b.y = bv.y;
            acc[t] = wmma_f32(a, b, acc[t]);
        }
    }

    // ---- epilogue: +ub, relu, +residual -> padded LDS tile ----
#pragma unroll
    for (int t = 0; t < 8; ++t) {
        const int col = t * 16 + ln16;
        const float bias = ub[col];
#pragma unroll
        for (int r = 0; r < 8; ++r) {
            const int M = r + 8 * kh;
            float v = acc[t][r] + bias;
            v = v > 0.f ? v : 0.f;
            v += hres[(size_t)(m0 + M) * H + col];
            ot[M * 132 + col] = v;
        }
    }
    asm volatile("s_wait_dscnt 0" ::: "memory");

    // ---- post-LN: 2 lanes per row (cols split 64/64), xor-16 combine ----
    const int rr = ln16;           // row within tile
    const int c0 = kh * 64;        // column base for this lane
    const float4* rp4 = (const float4*)(ot + rr * 132 + c0);
    float s = 0.f, s2 = 0.f;
#pragma unroll
    for (int i = 0; i < 16; ++i) {
        const float4 v = rp4[i];
        s  += v.x + v.y + v.z + v.w;
        s2 += v.x * v.x + v.y * v.y + v.z * v.z + v.w * v.w;
    }
    s  += __shfl_xor(s, 16, 32);
    s2 += __shfl_xor(s2, 16, 32);
    const float mean = s * (1.0f / H);
    const float var  = s2 * (1.0f / H) - mean * mean;
    const float rs   = rsqrtf(var + LN_EPS);

    float* orow = out + (size_t)(m0 + rr) * H + c0;
#pragma unroll
    for (int i = 0; i < 16; ++i) {
        const float4 v  = rp4[i];
        const float4 gg = ((const float4*)(pg + c0))[i];
        const float4 bb = ((const float4*)(pb + c0))[i];
        float4 o;
        o.x = (v.x - mean) * rs * gg.x + bb.x;
        o.y = (v.y - mean) * rs * gg.y + bb.y;
        o.z = (v.z - mean) * rs * gg.z + bb.z;
        o.w = (v.w - mean) * rs * gg.w + bb.w;
        ((float4*)orow)[i] = o;
    }
}

extern "C" void kernel_launch(void* const* d_in, const int* in_sizes, int n_in,
                              void* d_out, int out_size, void* d_ws, size_t ws_size,
                              hipStream_t stream) {
    const float* x   = (const float*)d_in[0];
    const int*   ei  = (const int*)  d_in[1];
    const float* dw  = (const float*)d_in[2];
    const float* db  = (const float*)d_in[3];
    const float* uw  = (const float*)d_in[4];
    const float* ub  = (const float*)d_in[5];
    const float* prg = (const float*)d_in[6];
    const float* prb = (const float*)d_in[7];
    const float* pog = (const float*)d_in[8];
    const float* pob = (const float*)d_in[9];

    float* out = (float*)d_out;         // also serves as agg accumulator
    float* h   = (float*)d_ws;          // pre-LN activations, N*H floats (8 MB)

    const int N = in_sizes[0] / H;      // 16384
    const int E = in_sizes[1] / 2;      // 524288

    // 1) pre-LN + zero agg (agg == d_out, re-zeroed every call)
    k_preln<<<N / 8, 256, 0, stream>>>(x, prg, prb, h, out);

    // 2) sparse adjacency aggregation via L2-resident f32 atomics
    const unsigned total = (unsigned)E * 32u;
    k_scatter<<<(total + 255u) / 256u, 256, 0, stream>>>(ei, h, out, E);

    // 3) fused WMMA MLP + relu + residual + post-LN (reads agg from d_out, overwrites it)
    k_mlp<<<N / 64, 128, 0, stream>>>(out, h, dw, db, uw, ub, pog, pob, out);
}